// BellmanOp_42563125903609
// MI455X (gfx1250) — compile-verified
//
#include <hip/hip_runtime.h>

// C51 categorical Bellman projection.
// Memory-bound: 412 MB traffic -> ~17.7 us floor at 23.3 TB/s HBM.
// Strategy: async-DMA a contiguous 160-row tile of probs into LDS
// (ASYNCcnt path, coalesced B128, non-temporal), per-row scatter-add in
// LDS via DS-indexed RMW, async-DMA the result tile back out (NT store).
// Streaming set (408 MB) > L2 (192 MB) with zero reuse -> NT hints keep
// L2 for atom_values + concurrent blocks instead of dead streaming lines.

#define NUM_ATOMS 51
#define ROWS      160                      // threads per block = rows per tile
#define TILE_EL   (ROWS * NUM_ATOMS)       // 8160 floats = 32640 B per buffer
#define V_MIN_F   (-10.0f)
#define V_MAX_F   (10.0f)
#define DELTA_F   (0.4f)                   // (10-(-10))/(51-1), fp32 as in reference

typedef __attribute__((address_space(3))) void lds_void_t;

__device__ __forceinline__ void async_ld_b128(const float* g, const float* l) {
    asm volatile("global_load_async_to_lds_b128 %0, %1, off th:TH_LOAD_NT"
                 :: "v"((lds_void_t*)l), "v"(g) : "memory");
}
__device__ __forceinline__ void async_ld_b32(const float* g, const float* l) {
    asm volatile("global_load_async_to_lds_b32 %0, %1, off th:TH_LOAD_NT"
                 :: "v"((lds_void_t*)l), "v"(g) : "memory");
}
__device__ __forceinline__ void async_st_b128(float* g, const float* l) {
    asm volatile("global_store_async_from_lds_b128 %0, %1, off th:TH_STORE_NT"
                 :: "v"(g), "v"((lds_void_t*)l) : "memory");
}
__device__ __forceinline__ void async_st_b32(float* g, const float* l) {
    asm volatile("global_store_async_from_lds_b32 %0, %1, off th:TH_STORE_NT"
                 :: "v"(g), "v"((lds_void_t*)l) : "memory");
}
__device__ __forceinline__ void wait_async0() {
    asm volatile("s_wait_asynccnt 0x0" ::: "memory");
}

__global__ void __launch_bounds__(ROWS)
c51_project_kernel(const float* __restrict__ reward,
                   const float* __restrict__ probs,
                   const float* __restrict__ atom_values,
                   float* __restrict__ out,
                   int bs)
{
    __shared__ __align__(16) float s_p[TILE_EL];   // staged probs tile
    __shared__ __align__(16) float s_o[TILE_EL];   // accumulated output tile

    const int        t       = threadIdx.x;
    const long long  rowBase = (long long)blockIdx.x * ROWS;
    const long long  base    = rowBase * NUM_ATOMS;        // flat element offset
    long long        remRows = (long long)bs - rowBase;
    const int        rowsHere = (remRows < ROWS) ? (int)remRows : ROWS;
    const int        elems   = rowsHere * NUM_ATOMS;       // 8160 for full tiles
    const int        nv4     = elems >> 2;                 // B128 chunk count

    // ---- Phase 1: async DMA probs tile (contiguous in global) into LDS ----
    const float* gsrc = probs + base;
    for (int c = t; c < nv4; c += ROWS)
        async_ld_b128(gsrc + 4 * c, &s_p[4 * c]);
    for (int c = 4 * nv4 + t; c < elems; c += ROWS)        // sub-16B tail (tail tile only)
        async_ld_b32(gsrc + c, &s_p[c]);

    // Zero the output tile concurrently with the in-flight async loads.
    for (int j = t; j < elems; j += ROWS)
        s_o[j] = 0.0f;

    wait_async0();          // my wave's async copies have landed in LDS
    __syncthreads();        // everyone's copies visible to everyone

    // ---- Phase 2: per-row projection, scatter-add into this row's LDS slice ----
    if (t < rowsHere) {
        const int   ro = t * NUM_ATOMS;
        const float r  = __builtin_nontemporal_load(&reward[rowBase + t]);
        for (int a = 0; a < NUM_ATOMS; ++a) {
            // Mirror the reference math exactly (fp32 clip / floor / ceil,
            // division by ATOM_DELTA, l==u -> coef 1.0 rule).
            float nv  = fminf(fmaxf(atom_values[a] + r, V_MIN_F), V_MAX_F);
            float idx = (nv - V_MIN_F) / DELTA_F;
            float lf  = floorf(idx);
            float uf  = ceilf(idx);
            int   li  = (int)lf;
            int   ui  = (int)uf;
            float pa  = s_p[ro + a];
            float lc  = (li == ui) ? 1.0f : (uf - idx);
            float uc  = idx - lf;
            // Single owner per row -> plain DS read-modify-write, no atomics.
            s_o[ro + li] += lc * pa;
            s_o[ro + ui] += uc * pa;
        }
    }
    __syncthreads();

    // ---- Phase 3: async DMA result tile LDS -> global (coalesced B128, NT) ----
    float* gdst = out + base;
    for (int c = t; c < nv4; c += ROWS)
        async_st_b128(gdst + 4 * c, &s_o[4 * c]);
    for (int c = 4 * nv4 + t; c < elems; c += ROWS)
        async_st_b32(gdst + c, &s_o[c]);
    wait_async0();          // stores issued into the memory hierarchy
}

extern "C" void kernel_launch(void* const* d_in, const int* in_sizes, int n_in,
                              void* d_out, int out_size, void* d_ws, size_t ws_size,
                              hipStream_t stream)
{
    const float* reward      = (const float*)d_in[0];
    const float* probs       = (const float*)d_in[1];
    const float* atom_values = (const float*)d_in[2];
    float*       out         = (float*)d_out;

    const int bs     = in_sizes[0];                 // 1,000,000
    const int blocks = (bs + ROWS - 1) / ROWS;      // 6250 (exact for BS=1e6)

    hipLaunchKernelGGL(c51_project_kernel, dim3(blocks), dim3(ROWS), 0, stream,
                       reward, probs, atom_values, out, bs);
}